// FDN_38079180046527
// MI455X (gfx1250) — compile-verified
//
#include <hip/hip_runtime.h>
#include <math.h>

// Problem constants
#define Bsz 4096
#define Dd  256
#define Hh  2048
#define Oo  256

typedef __attribute__((ext_vector_type(2))) float v2f;
typedef __attribute__((ext_vector_type(8))) float v8f;

// A-tile LDS swizzle: 128 rows x 32 cols (8 chunks of 4 floats), chunk XOR'ed by row.
// 16B-aligned contiguous chunks (async B128 OK); <=2-way bank conflicts on 64 banks.
#define ASWZ(r, chunk) (((r) * 32) + ((((chunk) ^ ((r) & 7))) << 2))

struct ExpertArgs {
    const float* xp[3];   // A feature parts, each [B x 256] row-major
    const float* W;       // [K x H] row-major base (per-z stride wstride)
    const float* bias;    // [H] base (per-z stride H)
    float*       out;     // [B x H] base (per-z stride outstride)
    int  nparts;          // K = nparts * 256
    long wstride;
    long outstride;
};

struct HeadArgs {
    const float* Hsrc[4]; // [B x 2048] hidden activations
    const float* gate[4]; // [B] per-row gates
    const float* W[4];    // [2048 x 256]
    const float* bias[4]; // [256]
    float*       out;     // [B x 256]
    int ncontrib;
};

// Low 32 bits of a flat LDS address == byte offset within the group's LDS allocation.
__device__ __forceinline__ unsigned lds_off(const void* p)
{
    return (unsigned)(unsigned long long)p;
}

// Async DMA: 16 bytes global -> LDS, tracked by ASYNCcnt (GVS addressing mode).
__device__ __forceinline__ void async_copy16(unsigned ldsByteOff, unsigned gByteOff,
                                             const void* base)
{
    asm volatile("global_load_async_to_lds_b128 %0, %1, %2"
                 :: "v"(ldsByteOff), "v"(gByteOff), "s"(base)
                 : "memory");
}

// Issue one 128x32 A tile + one 32x128 B tile: exactly 8 async B128 ops per thread.
__device__ __forceinline__ void issue_tile(unsigned asb, unsigned bsb,
                                           const float* Ap, int lda, int kc,
                                           const float* Wp, int ldb, int k0,
                                           int m0, int n0, int tid)
{
#pragma unroll
    for (int i = 0; i < 4; ++i) {               // A tile: 128 rows x 8 chunks
        const int idx = tid + i * 256;
        const int r   = idx >> 3;
        const int cq  = idx & 7;
        const unsigned lofs = asb + (unsigned)(ASWZ(r, cq) * 4);
        const unsigned gofs = (unsigned)(((m0 + r) * lda + kc + cq * 4) * 4);
        async_copy16(lofs, gofs, Ap);
    }
#pragma unroll
    for (int i = 0; i < 4; ++i) {               // B tile: 32 rows x 128 cols
        const int idx = tid + i * 256;
        const int r   = idx >> 5;
        const int q   = (idx & 31) << 2;
        const unsigned lofs = bsb + (unsigned)((r * 128 + q) * 4);
        const unsigned gofs = (unsigned)(((long)(k0 + r) * ldb + n0 + q) * 4);
        async_copy16(lofs, gofs, Wp);
    }
}

// One K-block (32) of WMMA: 8 K-steps x (2 M-tiles x 4 N-tiles).
// GATED: scale A fragments by per-row gates (row scaling of A == gating the GEMM).
template <bool GATED>
__device__ __forceinline__ void wmma_block(const float* As, const float* Bs,
                                           v8f (&acc)[2][4], int wm, int wn,
                                           int l15, int lhalf, float g0, float g1)
{
#pragma unroll
    for (int kk = 0; kk < 8; ++kk) {
        v2f a[2], b[4];
#pragma unroll
        for (int mt = 0; mt < 2; ++mt) {
            const int row = wm + mt * 16 + l15;
            const int off = ASWZ(row, kk) + 2 * lhalf;   // both words in same chunk
            a[mt].x = As[off];
            a[mt].y = As[off + 1];
        }
        if (GATED) {
            a[0].x *= g0; a[0].y *= g0;
            a[1].x *= g1; a[1].y *= g1;
        }
        const int kb = kk * 4 + 2 * lhalf;
#pragma unroll
        for (int nt = 0; nt < 4; ++nt) {
            const int col = wn + nt * 16 + l15;
            b[nt].x = Bs[kb * 128 + col];
            b[nt].y = Bs[(kb + 1) * 128 + col];
        }
#pragma unroll
        for (int mt = 0; mt < 2; ++mt)
#pragma unroll
            for (int nt = 0; nt < 4; ++nt)
                acc[mt][nt] = __builtin_amdgcn_wmma_f32_16x16x4_f32(
                    false, a[mt], false, b[nt], (short)0, acc[mt][nt], false, false);
    }
}

// C = relu(concat(xp...) @ W + bias), C is [B x 2048]; grid.z batches experts.
__global__ __launch_bounds__(256) void expert_kernel(ExpertArgs arg)
{
    __shared__ float As[2][128 * 32];
    __shared__ float Bs[2][32 * 128];

    const int tid   = threadIdx.x;
    const int lane  = tid & 31;
    const int wid   = tid >> 5;
    const int wm    = (wid & 3) * 32;
    const int wn    = (wid >> 2) * 64;
    const int l15   = lane & 15;
    const int lhalf = lane >> 4;

    const int z = blockIdx.z;
    const float* W    = arg.W    + (long)z * arg.wstride;
    const float* bias = arg.bias + (long)z * Hh;
    float*       out  = arg.out  + (long)z * arg.outstride;
    const int n0 = blockIdx.x * 128;
    const int m0 = blockIdx.y * 128;
    const int nb = arg.nparts * (Dd / 32);      // K blocks of 32

    const unsigned asB[2] = { lds_off(&As[0][0]), lds_off(&As[1][0]) };
    const unsigned bsB[2] = { lds_off(&Bs[0][0]), lds_off(&Bs[1][0]) };

    const v8f zero = {0.f, 0.f, 0.f, 0.f, 0.f, 0.f, 0.f, 0.f};
    v8f acc[2][4];
#pragma unroll
    for (int mt = 0; mt < 2; ++mt)
#pragma unroll
        for (int nt = 0; nt < 4; ++nt) acc[mt][nt] = zero;

    // Software-pipelined, double-buffered async main loop.
    issue_tile(asB[0], bsB[0], arg.xp[0], Dd, 0, W, Hh, 0, m0, n0, tid);
    for (int j = 0; j < nb; ++j) {
        if (j + 1 < nb) {
            const int k0n = (j + 1) * 32;
            issue_tile(asB[(j + 1) & 1], bsB[(j + 1) & 1],
                       arg.xp[k0n >> 8], Dd, k0n & (Dd - 1), W, Hh, k0n, m0, n0, tid);
            asm volatile("s_wait_asynccnt 0x8" ::: "memory");  // block j landed
        } else {
            asm volatile("s_wait_asynccnt 0x0" ::: "memory");
        }
        __syncthreads();
        wmma_block<false>(&As[j & 1][0], &Bs[j & 1][0], acc, wm, wn, l15, lhalf, 1.f, 1.f);
        __syncthreads();
    }

#pragma unroll
    for (int mt = 0; mt < 2; ++mt) {
#pragma unroll
        for (int nt = 0; nt < 4; ++nt) {
            const int col = n0 + wn + nt * 16 + l15;
            const float bv = bias[col];
#pragma unroll
            for (int r = 0; r < 8; ++r) {
                const int row = m0 + wm + mt * 16 + r + 8 * lhalf;
                const float v = acc[mt][nt][r] + bv;
                out[(long)row * Hh + col] = v > 0.f ? v : 0.f;
            }
        }
    }
}

// out = sum_c gate_c[b] * (H_c @ W_c) + sum_c bias_c   (out is [B x 256])
__global__ __launch_bounds__(256) void head_kernel(HeadArgs arg)
{
    __shared__ float As[2][128 * 32];
    __shared__ float Bs[2][32 * 128];

    const int tid   = threadIdx.x;
    const int lane  = tid & 31;
    const int wid   = tid >> 5;
    const int wm    = (wid & 3) * 32;
    const int wn    = (wid >> 2) * 64;
    const int l15   = lane & 15;
    const int lhalf = lane >> 4;

    const int n0 = blockIdx.x * 128;
    const int m0 = blockIdx.y * 128;
    const int nb = arg.ncontrib * (Hh / 32);

    const unsigned asB[2] = { lds_off(&As[0][0]), lds_off(&As[1][0]) };
    const unsigned bsB[2] = { lds_off(&Bs[0][0]), lds_off(&Bs[1][0]) };

    const v8f zero = {0.f, 0.f, 0.f, 0.f, 0.f, 0.f, 0.f, 0.f};
    v8f acc[2][4];
#pragma unroll
    for (int mt = 0; mt < 2; ++mt)
#pragma unroll
        for (int nt = 0; nt < 4; ++nt) acc[mt][nt] = zero;

    issue_tile(asB[0], bsB[0], arg.Hsrc[0], Hh, 0, arg.W[0], Oo, 0, m0, n0, tid);
    for (int j = 0; j < nb; ++j) {
        if (j + 1 < nb) {
            const int cn  = (j + 1) >> 6;
            const int k0n = ((j + 1) & 63) * 32;
            issue_tile(asB[(j + 1) & 1], bsB[(j + 1) & 1],
                       arg.Hsrc[cn], Hh, k0n, arg.W[cn], Oo, k0n, m0, n0, tid);
            asm volatile("s_wait_asynccnt 0x8" ::: "memory");
        } else {
            asm volatile("s_wait_asynccnt 0x0" ::: "memory");
        }
        __syncthreads();
        const int c = j >> 6;
        const float g0 = arg.gate[c][m0 + wm + l15];        // rows of M-tile 0
        const float g1 = arg.gate[c][m0 + wm + 16 + l15];   // rows of M-tile 1
        wmma_block<true>(&As[j & 1][0], &Bs[j & 1][0], acc, wm, wn, l15, lhalf, g0, g1);
        __syncthreads();
    }

#pragma unroll
    for (int mt = 0; mt < 2; ++mt) {
#pragma unroll
        for (int nt = 0; nt < 4; ++nt) {
            const int col = n0 + wn + nt * 16 + l15;
            float bv = 0.f;
            for (int c = 0; c < arg.ncontrib; ++c) bv += arg.bias[c][col];
#pragma unroll
            for (int r = 0; r < 8; ++r) {
                const int row = m0 + wm + mt * 16 + r + 8 * lhalf;
                arg.out[(long)row * Oo + col] = acc[mt][nt][r] + bv;
            }
        }
    }
}

// g[z,b] = sigmoid(dot(src[z,b,:K], wg[z,:K]) + bg[z]); one wave32 per row.
__global__ __launch_bounds__(256) void gate_kernel(const float* src, long src_zstride,
                                                   const float* wg, long wg_zstride,
                                                   const float* bg,
                                                   float* g, long g_zstride, int K)
{
    const int lane = threadIdx.x & 31;
    const int wid  = threadIdx.x >> 5;
    const int z = blockIdx.y;
    const int b = blockIdx.x * 8 + wid;
    const float* s = src + (long)z * src_zstride + (long)b * K;
    const float* w = wg + (long)z * wg_zstride;
    float sum = 0.f;
    for (int k = lane; k < K; k += 32) sum += s[k] * w[k];
#pragma unroll
    for (int off = 16; off > 0; off >>= 1) sum += __shfl_xor(sum, off, 32);
    if (lane == 0) {
        const float t = sum + bg[z];
        g[(long)z * g_zstride + b] = 1.f / (1.f + __expf(-t));
    }
}

extern "C" void kernel_launch(void* const* d_in, const int* in_sizes, int n_in,
                              void* d_out, int out_size, void* d_ws, size_t ws_size,
                              hipStream_t stream)
{
    (void)in_sizes; (void)n_in; (void)out_size; (void)ws_size;

    const float* x[8];
    for (int i = 0; i < 8; ++i) x[i] = (const float*)d_in[i];
    const float* W_task  = (const float*)d_in[8];
    const float* b_task  = (const float*)d_in[9];
    const float* Wg_task = (const float*)d_in[10];
    const float* bg_task = (const float*)d_in[11];
    const float* Wo_task = (const float*)d_in[12];
    const float* bo_task = (const float*)d_in[13];
    const float* W_b2    = (const float*)d_in[14];
    const float* b_b2    = (const float*)d_in[15];
    const float* Wg_b2   = (const float*)d_in[16];
    const float* bg_b2   = (const float*)d_in[17];
    const float* Wo_b2   = (const float*)d_in[18];
    const float* bo_b2   = (const float*)d_in[19];
    const float* W_b3    = (const float*)d_in[20];
    const float* b_b3    = (const float*)d_in[21];
    const float* Wg_b3   = (const float*)d_in[22];
    const float* bg_b3   = (const float*)d_in[23];
    const float* Wo_b3   = (const float*)d_in[24];
    const float* bo_b3   = (const float*)d_in[25];

    // Output layout: task_out [8,B,O] ++ se2 [4,B,H] ++ se3 [6,B,H]
    float* task_out = (float*)d_out;
    float* se2 = task_out + (long)8 * Bsz * Oo;
    float* se3 = se2 + (long)4 * Bsz * Hh;

    // Workspace: te [B,H] (reused per task) + gates
    float* te  = (float*)d_ws;
    float* g_t = te + (long)Bsz * Hh;
    float* g2  = g_t + Bsz;
    float* g3  = g2 + (long)4 * Bsz;

    const dim3 blk(256);

    // Shared branch-2 experts -> se2 (in d_out), z = expert
    {
        ExpertArgs a{};
        a.xp[0] = x[2]; a.xp[1] = x[3]; a.xp[2] = x[3];
        a.W = W_b2; a.bias = b_b2; a.out = se2;
        a.nparts = 2; a.wstride = (long)2 * Dd * Hh; a.outstride = (long)Bsz * Hh;
        expert_kernel<<<dim3(Hh / 128, Bsz / 128, 4), blk, 0, stream>>>(a);
    }
    // Shared branch-3 experts -> se3
    {
        ExpertArgs a{};
        a.xp[0] = x[5]; a.xp[1] = x[6]; a.xp[2] = x[7];
        a.W = W_b3; a.bias = b_b3; a.out = se3;
        a.nparts = 3; a.wstride = (long)3 * Dd * Hh; a.outstride = (long)Bsz * Hh;
        expert_kernel<<<dim3(Hh / 128, Bsz / 128, 6), blk, 0, stream>>>(a);
    }
    // Gates on hidden activations
    gate_kernel<<<dim3(Bsz / 8, 4), blk, 0, stream>>>(se2, (long)Bsz * Hh, Wg_b2, Hh, bg_b2, g2, Bsz, Hh);
    gate_kernel<<<dim3(Bsz / 8, 6), blk, 0, stream>>>(se3, (long)Bsz * Hh, Wg_b3, Hh, bg_b3, g3, Bsz, Hh);

    for (int t = 0; t < 8; ++t) {
        // Task expert -> te (scratch, reused serially on the stream)
        ExpertArgs a{};
        a.xp[0] = x[t]; a.xp[1] = x[t]; a.xp[2] = x[t];
        a.W = W_task + (long)t * Dd * Hh; a.bias = b_task + (long)t * Hh; a.out = te;
        a.nparts = 1; a.wstride = 0; a.outstride = 0;
        expert_kernel<<<dim3(Hh / 128, Bsz / 128, 1), blk, 0, stream>>>(a);

        // Task gate on x[t]
        gate_kernel<<<dim3(Bsz / 8, 1), blk, 0, stream>>>(x[t], 0, Wg_task + (long)t * Dd, 0,
                                                          bg_task + t, g_t, 0, Dd);

        // Head: task contribution + routed shared-expert contributions
        HeadArgs h{};
        int nc = 0;
        h.Hsrc[nc] = te; h.gate[nc] = g_t;
        h.W[nc] = Wo_task + (long)t * Hh * Oo; h.bias[nc] = bo_task + (long)t * Oo; nc++;

        auto add2 = [&](int e) {
            h.Hsrc[nc] = se2 + (long)e * Bsz * Hh; h.gate[nc] = g2 + (long)e * Bsz;
            h.W[nc] = Wo_b2 + (long)e * Hh * Oo;   h.bias[nc] = bo_b2 + (long)e * Oo; nc++;
        };
        auto add3 = [&](int e) {
            h.Hsrc[nc] = se3 + (long)e * Bsz * Hh; h.gate[nc] = g3 + (long)e * Bsz;
            h.W[nc] = Wo_b3 + (long)e * Hh * Oo;   h.bias[nc] = bo_b3 + (long)e * Oo; nc++;
        };
        if (t == 1) { add2(0); add2(1); }
        else if (t == 2) { add2(2); }
        else if (t == 3) { add2(3); }
        else if (t == 4) { add3(0); add3(1); add3(2); }
        else if (t == 5) { add3(3); }
        else if (t == 6) { add3(4); }
        else if (t == 7) { add3(5); }
        h.ncontrib = nc;
        h.out = task_out + (long)t * Bsz * Oo;
        for (int c = nc; c < 4; ++c) { // keep unused slots valid (never read)
            h.Hsrc[c] = te; h.gate[c] = g_t; h.W[c] = Wo_task; h.bias[c] = bo_task;
        }
        head_kernel<<<dim3(Oo / 128, Bsz / 128, 1), blk, 0, stream>>>(h);
    }
}